// AlphaModel_78658031059186
// MI455X (gfx1250) — compile-verified
//
#include <hip/hip_runtime.h>
#include <cstdint>

#define NRELS 64

typedef float fx4 __attribute__((ext_vector_type(4)));
typedef int   ix4 __attribute__((ext_vector_type(4)));

// Exact d=3 sparsemax (Martins & Astudillo), faithful to the reference's
// support test "1 + k*zs > cs" with ksup = sum(support).
__device__ __forceinline__ void sparsemax3(float a, float b, float c,
                                           float& oa, float& ob, float& oc) {
  float hi  = fmaxf(a, b);
  float lo  = fminf(a, b);
  float zs0 = fmaxf(hi, c);              // max
  float zs2 = fminf(lo, c);              // min
  float zs1 = fmaxf(lo, fminf(hi, c));   // median
  float cs2 = zs0 + zs1;
  float cs3 = cs2 + zs2;
  int s2 = (1.0f + 2.0f * zs1) > cs2;
  int s3 = (1.0f + 3.0f * zs2) > cs3;
  int k = 1 + s2 + s3;
  float cs = (k == 3) ? cs3 : ((k == 2) ? cs2 : zs0);
  float rk = (k == 3) ? (1.0f / 3.0f) : ((k == 2) ? 0.5f : 1.0f);
  float tau = (cs - 1.0f) * rk;
  oa = fmaxf(a - tau, 0.0f);
  ob = fmaxf(b - tau, 0.0f);
  oc = fmaxf(c - tau, 0.0f);
}

// One edge: tr -> 12 floats in LDS: M[rel] row-major (9) then beta[rel] (3).
__device__ __forceinline__ void alpha_edge(const float* __restrict__ tr,
                                           float chx, float chy, float chz,
                                           float px, float py, float pz,
                                           float eps, float sf,
                                           float& ox, float& oy, float& oz) {
  const fx4* mt = (const fx4*)tr;        // 3 x ds_load_b128
  fx4 m0 = mt[0], m1 = mt[1], m2 = mt[2];
  // c_i = sum_j M[i][j] * child[j]
  float cx = m0.x * chx + m0.y * chy + m0.z * chz;
  float cy = m0.w * chx + m1.x * chy + m1.y * chz;
  float cz = m1.z * chx + m1.w * chy + m2.x * chz;
  float bx = m2.y, by = m2.z, bz = m2.w; // beta[rel]
  sparsemax3(cx, cy, cz, cx, cy, cz);
  sparsemax3(cx, cy, cz, cx, cy, cz);    // reference re-applies sparsemax
  sparsemax3(px, py, pz, px, py, pz);
  float ax = (1.0f - bx) * px + bx * cx;
  float ay = (1.0f - by) * py + by * cy;
  float az = (1.0f - bz) * pz + bz * cz;
  // scale(): entropy of clamped mixture + cosine similarity
  float zx = fmaxf(px + cx, eps);
  float zy = fmaxf(py + cy, eps);
  float zz = fmaxf(pz + cz, eps);
  float rs = 1.0f / (zx + zy + zz);
  zx *= rs; zy *= rs; zz *= rs;
  float H = -(zx * __logf(zx) + zy * __logf(zy) + zz * __logf(zz));
  float dotpc = px * cx + py * cy + pz * cz;
  float np = sqrtf(px * px + py * py + pz * pz);
  float nc = sqrtf(cx * cx + cy * cy + cz * cz);
  float cosv = 0.1f + dotpc / fmaxf(np * nc, 1e-10f);
  float scale = sf * cosv / H;
  ox = fmaxf(ax * scale, 0.001f);
  oy = fmaxf(ay * scale, 0.001f);
  oz = fmaxf(az * scale, 0.001f);
}

__global__ __launch_bounds__(256) void alpha_kernel(
    const float* __restrict__ prnt, const float* __restrict__ child,
    const float* __restrict__ Mg, const float* __restrict__ betag,
    const float* __restrict__ epsp, const float* __restrict__ sfp,
    const int* __restrict__ rels, float* __restrict__ out, int n) {
  __shared__ __align__(16) float stage[NRELS * 12]; // raw: M (64*9) then beta (64*3)
  __shared__ __align__(16) float tbl[NRELS * 12];   // interleaved per-rel [M(9)|beta(3)]

  // ---- Stage the 3 KB parameter table into LDS via CDNA5 async DMA ----
  if (threadIdx.x < 32) {                 // one full wave32, EXEC all ones
    const uint32_t lane = threadIdx.x;
    const uint32_t lds0 = (uint32_t)(uintptr_t)(&stage[0]); // low 32b = LDS offset
    // M: 64*9*4 = 2304 B = 144 x b128
    for (uint32_t i = lane; i < (NRELS * 9u * 4u) / 16u; i += 32u) {
      uint32_t ldsa = lds0 + i * 16u;
      const void* g = (const char*)Mg + (size_t)i * 16u;
      asm volatile("global_load_async_to_lds_b128 %0, %1, off"
                   :: "v"(ldsa), "v"(g) : "memory");
    }
    // beta: 64*3*4 = 768 B = 48 x b128
    for (uint32_t i = lane; i < (NRELS * 3u * 4u) / 16u; i += 32u) {
      uint32_t ldsa = lds0 + (uint32_t)(NRELS * 9 * 4) + i * 16u;
      const void* g = (const char*)betag + (size_t)i * 16u;
      asm volatile("global_load_async_to_lds_b128 %0, %1, off"
                   :: "v"(ldsa), "v"(g) : "memory");
    }
    asm volatile("s_wait_asynccnt 0" ::: "memory");
  }
  __syncthreads();
  // Interleave so each rel's 12 floats form one 48 B, 16 B-aligned record
  // -> per-edge gather is exactly 3 x ds_load_b128.
  for (int idx = threadIdx.x; idx < NRELS * 12; idx += (int)blockDim.x) {
    int rel = idx / 12, j = idx - rel * 12;
    tbl[idx] = (j < 9) ? stage[rel * 9 + j]
                       : stage[NRELS * 9 + rel * 3 + (j - 9)];
  }
  __syncthreads();

  const float eps = epsp[0];   // uniform -> s_load
  const float sf  = sfp[0];

  int i4 = (blockIdx.x * (int)blockDim.x + (int)threadIdx.x) * 4;
  if (i4 >= n) return;

  if (i4 + 4 <= n) {
    // 4 edges / thread: every stream access is a 128-bit non-temporal op.
    const fx4* chv = (const fx4*)(child + (size_t)i4 * 3);
    const fx4* prv = (const fx4*)(prnt + (size_t)i4 * 3);
    fx4 c0 = __builtin_nontemporal_load(chv + 0);
    fx4 c1 = __builtin_nontemporal_load(chv + 1);
    fx4 c2 = __builtin_nontemporal_load(chv + 2);
    fx4 p0 = __builtin_nontemporal_load(prv + 0);
    fx4 p1 = __builtin_nontemporal_load(prv + 1);
    fx4 p2 = __builtin_nontemporal_load(prv + 2);
    ix4 rr = __builtin_nontemporal_load((const ix4*)(rels + i4));

    float ch[12], pr[12], ov[12];
    *(fx4*)(ch + 0) = c0; *(fx4*)(ch + 4) = c1; *(fx4*)(ch + 8) = c2;
    *(fx4*)(pr + 0) = p0; *(fx4*)(pr + 4) = p1; *(fx4*)(pr + 8) = p2;
    int rA[4];
    *(ix4*)rA = rr;

#pragma unroll
    for (int e = 0; e < 4; ++e) {
      const float* tr = &tbl[rA[e] * 12];
      alpha_edge(tr,
                 ch[3 * e + 0], ch[3 * e + 1], ch[3 * e + 2],
                 pr[3 * e + 0], pr[3 * e + 1], pr[3 * e + 2],
                 eps, sf,
                 ov[3 * e + 0], ov[3 * e + 1], ov[3 * e + 2]);
    }
    fx4* op = (fx4*)(out + (size_t)i4 * 3);
    __builtin_nontemporal_store(*(fx4*)(ov + 0), op + 0);
    __builtin_nontemporal_store(*(fx4*)(ov + 4), op + 1);
    __builtin_nontemporal_store(*(fx4*)(ov + 8), op + 2);
  } else {
    // tail (N % 4 != 0)
    for (int i = i4; i < n; ++i) {
      const float* tr = &tbl[rels[i] * 12];
      float ox, oy, oz;
      alpha_edge(tr,
                 child[3 * i + 0], child[3 * i + 1], child[3 * i + 2],
                 prnt[3 * i + 0], prnt[3 * i + 1], prnt[3 * i + 2],
                 eps, sf, ox, oy, oz);
      out[3 * i + 0] = ox; out[3 * i + 1] = oy; out[3 * i + 2] = oz;
    }
  }
}

extern "C" void kernel_launch(void* const* d_in, const int* in_sizes, int n_in,
                              void* d_out, int out_size, void* d_ws, size_t ws_size,
                              hipStream_t stream) {
  // setup_inputs order: prnt_probs, child_probs, M, beta, z_epsilon,
  //                     scale_factor, rels, var_sfx
  const float* prnt  = (const float*)d_in[0];
  const float* child = (const float*)d_in[1];
  const float* Mg    = (const float*)d_in[2];
  const float* betag = (const float*)d_in[3];
  const float* epsp  = (const float*)d_in[4];
  const float* sfp   = (const float*)d_in[5];
  const int*   rels  = (const int*)d_in[6];
  float* out = (float*)d_out;

  const int n = in_sizes[6];              // rels has N elements
  if (n <= 0) return;
  const int threads_total = (n + 3) / 4;  // 4 edges per thread
  const int block = 256;                  // 8 wave32s per workgroup
  const int grid = (threads_total + block - 1) / block;
  alpha_kernel<<<grid, block, 0, stream>>>(prnt, child, Mg, betag,
                                           epsp, sfp, rels, out, n);
}